// Net_31516470018161
// MI455X (gfx1250) — compile-verified
//
#include <hip/hip_runtime.h>
#include <math.h>

// ---------------- problem constants (from reference) ----------------
#define NN 16384      // nodes
#define EE 262144     // edges
#define GG 64         // graphs
#define CD 256        // channels (heads*d) for all layers
#define INC 128       // input channels

typedef float v2f __attribute__((ext_vector_type(2)));
typedef float v8f __attribute__((ext_vector_type(8)));

// ---------------- helpers: order-preserving float<->uint ----------------
__device__ __forceinline__ unsigned flipF(float f) {
    unsigned u = __float_as_uint(f);
    return (u & 0x80000000u) ? ~u : (u | 0x80000000u);
}
__device__ __forceinline__ float unflipF(unsigned u) {
    u = (u & 0x80000000u) ? (u & 0x7FFFFFFFu) : ~u;
    return __uint_as_float(u);
}
#define FLIP_NEG_INF 0x007FFFFFu   // flipF(-inf)

// ---------------- fill kernels ----------------
__global__ void fill_f32(float* __restrict__ p, float v, size_t n) {
    size_t i = (size_t)blockIdx.x * blockDim.x + threadIdx.x;
    if (i < n) p[i] = v;
}
__global__ void fill_u32(unsigned* __restrict__ p, unsigned v, size_t n) {
    size_t i = (size_t)blockIdx.x * blockDim.x + threadIdx.x;
    if (i < n) p[i] = v;
}

// ---------------- fp32 WMMA GEMM: C[M,Ncol] = A[M,K] @ W[K,Ncol] + bias ----------------
// One wave computes one 16x16 output tile via V_WMMA_F32_16X16X4_F32 (K-step 4).
// A tile layout (ISA 7.12.2, 32-bit A 16x4): lanes 0-15 -> row=lane, K=k..k+1;
// lanes 16-31 -> row=lane-16, K=k+2..k+3. C/D: vgpr r -> row r (+8 for upper half),
// col = lane&15.
__global__ void __launch_bounds__(256)
gemm_wmma_f32(const float* __restrict__ A, const float* __restrict__ W,
              const float* __restrict__ bias, float* __restrict__ C,
              int Kdim, int Ncol) {
    const int wave = threadIdx.x >> 5;
    const int lane = threadIdx.x & 31;
    const int half = lane >> 4;
    const int l16  = lane & 15;
    const int ntile = Ncol >> 4;
    int tile = blockIdx.x * 8 + wave;
    int m0 = (tile / ntile) << 4;
    int n0 = (tile % ntile) << 4;

    v8f acc = {};
    const float* Arow = A + (size_t)(m0 + l16) * Kdim;
    for (int k = 0; k < Kdim; k += 4) {
        int ak = k + half * 2;
        v2f a, b;
        a.x = Arow[ak];
        a.y = Arow[ak + 1];
        b.x = W[(size_t)ak * Ncol + n0 + l16];
        b.y = W[(size_t)(ak + 1) * Ncol + n0 + l16];
        acc = __builtin_amdgcn_wmma_f32_16x16x4_f32(false, a, false, b,
                                                    (short)0, acc, false, false);
    }
    float bv = bias[n0 + l16];
#pragma unroll
    for (int r = 0; r < 8; ++r) {
        int row = m0 + r + half * 8;
        C[(size_t)row * Ncol + n0 + l16] = acc[r] + bv;
    }
}

// ---------------- edge logits: per-edge attention scores ----------------
// block = one edge, 256 threads = channels. e_c = ea@We recomputed on the fly.
__global__ void __launch_bounds__(256)
edge_logits_kernel(const float* __restrict__ Q, const float* __restrict__ Kf,
                   const float* __restrict__ ea, const float* __restrict__ We,
                   const int* __restrict__ src, const int* __restrict__ dst,
                   float* __restrict__ logits, unsigned* __restrict__ mflip,
                   int heads, int d, float inv_sqrt_d) {
    int e = blockIdx.x;
    int c = threadIdx.x;
    int s = src[e], t = dst[e];
    float ev = 0.f;
#pragma unroll
    for (int j = 0; j < 4; ++j) ev += ea[(size_t)e * 4 + j] * We[j * CD + c];
    float prod = Q[(size_t)t * CD + c] * (Kf[(size_t)s * CD + c] + ev);

    __shared__ float red[256];
    red[c] = prod;
    __syncthreads();
    for (int off = d >> 1; off > 0; off >>= 1) {
        if ((c & (d - 1)) < off) red[c] += red[c + off];
        __syncthreads();
    }
    if ((c & (d - 1)) == 0) {
        int h = c / d;
        float lg = red[c] * inv_sqrt_d;
        logits[(size_t)e * heads + h] = lg;
        atomicMax(&mflip[t * heads + h], flipF(lg));
    }
}

// ---------------- softmax numerator + segment sum ----------------
__global__ void edge_softmax_kernel(float* __restrict__ lp, const unsigned* __restrict__ mflip,
                                    float* __restrict__ ssum, const int* __restrict__ dst,
                                    int heads, int total) {
    int i = blockIdx.x * blockDim.x + threadIdx.x;
    if (i >= total) return;
    int e = i / heads, h = i - e * heads;
    int t = dst[e];
    float m = unflipF(mflip[t * heads + h]);
    float p = __expf(lp[i] - m);
    lp[i] = p;
    atomicAdd(&ssum[t * heads + h], p);
}

// ---------------- weighted aggregation: agg[dst] += alpha * (v[src] + e) ----------------
__global__ void __launch_bounds__(256)
edge_aggregate_kernel(const float* __restrict__ p, const float* __restrict__ ssum,
                      const float* __restrict__ V, const float* __restrict__ ea,
                      const float* __restrict__ We, const int* __restrict__ src,
                      const int* __restrict__ dst, float* __restrict__ agg,
                      int heads, int dshift) {
    int e = blockIdx.x;
    int c = threadIdx.x;
    int s = src[e], t = dst[e];
    int h = c >> dshift;
    float alpha = p[(size_t)e * heads + h] / ssum[t * heads + h];
    float ev = 0.f;
#pragma unroll
    for (int j = 0; j < 4; ++j) ev += ea[(size_t)e * 4 + j] * We[j * CD + c];
    atomicAdd(&agg[(size_t)t * CD + c], alpha * (V[(size_t)s * CD + c] + ev));
}

// ---------------- skip connection + BN column statistics ----------------
__global__ void __launch_bounds__(256)
skip_stats_kernel(float* __restrict__ y, const float* __restrict__ S,
                  float* __restrict__ colsum, float* __restrict__ colsumsq,
                  int rows_per_block) {
    int c = threadIdx.x;
    int n0 = blockIdx.x * rows_per_block;
    float sm = 0.f, sq = 0.f;
    for (int r = 0; r < rows_per_block; ++r) {
        size_t idx = (size_t)(n0 + r) * CD + c;
        float v = y[idx] + S[idx];
        y[idx] = v;
        sm += v;
        sq += v * v;
    }
    atomicAdd(&colsum[c], sm);
    atomicAdd(&colsumsq[c], sq);
}

__global__ void bn_apply_kernel(const float* __restrict__ y, const float* __restrict__ colsum,
                                const float* __restrict__ colsumsq, const float* __restrict__ g,
                                const float* __restrict__ b, float* __restrict__ out) {
    int i = blockIdx.x * 256 + threadIdx.x;   // covers NN*CD
    int c = i & (CD - 1);
    const float invn = 1.0f / (float)NN;
    float mu = colsum[c] * invn;
    float var = colsumsq[c] * invn - mu * mu;
    out[i] = (y[i] - mu) * rsqrtf(var + 1e-5f) * g[c] + b[c];
}

// ---------------- graph pooling ----------------
__global__ void __launch_bounds__(256)
pool_scatter_kernel(const float* __restrict__ x, const int* __restrict__ batch,
                    float* __restrict__ padd, unsigned* __restrict__ pmax,
                    float* __restrict__ counts) {
    int n = blockIdx.x;
    int c = threadIdx.x;
    int g = batch[n];
    float v = x[(size_t)n * CD + c];
    atomicAdd(&padd[g * CD + c], v);
    atomicMax(&pmax[g * CD + c], flipF(v));
    if (c == 0) atomicAdd(&counts[g], 1.0f);
}

__global__ void pool_finalize_kernel(const float* __restrict__ padd, const unsigned* __restrict__ pmax,
                                     const float* __restrict__ counts, float* __restrict__ h0) {
    int i = blockIdx.x * 256 + threadIdx.x;   // GG*CD
    int g = i >> 8, c = i & 255;
    float cnt = counts[g];
    float add = padd[i];
    float mx = (cnt > 0.f) ? unflipF(pmax[i]) : 0.f;
    h0[(size_t)g * 768 + c]       = add;
    h0[(size_t)g * 768 + 256 + c] = mx;
    h0[(size_t)g * 768 + 512 + c] = add / fmaxf(cnt, 1.0f);
}

// ---------------- MLP head ----------------
__global__ void mlp1_kernel(const float* __restrict__ h0, const float* __restrict__ W,
                            const float* __restrict__ b, const float* __restrict__ pa,
                            float* __restrict__ h1) {
    int i = blockIdx.x * blockDim.x + threadIdx.x;   // GG*768
    int g = i / 768, j = i - g * 768;
    const float* row = h0 + (size_t)g * 768;
    float acc = b[j];
    for (int k = 0; k < 768; ++k) acc += row[k] * W[(size_t)k * 768 + j];
    float a = pa[0];
    h1[i] = acc > 0.f ? acc : a * acc;
}

__global__ void mlp2_kernel(const float* __restrict__ h1, const float* __restrict__ W,
                            const float* __restrict__ b, float* __restrict__ out) {
    int g = blockIdx.x * blockDim.x + threadIdx.x;
    if (g >= GG) return;
    const float* row = h1 + (size_t)g * 768;
    float z0 = b[0], z1 = b[1];
    for (int k = 0; k < 768; ++k) {
        float v = row[k];
        z0 += v * W[k * 2 + 0];
        z1 += v * W[k * 2 + 1];
    }
    float m = fmaxf(z0, z1);
    float lse = m + logf(expf(z0 - m) + expf(z1 - m));
    out[g * 2 + 0] = z0 - lse;
    out[g * 2 + 1] = z1 - lse;
}

// ---------------- host orchestration ----------------
extern "C" void kernel_launch(void* const* d_in, const int* in_sizes, int n_in,
                              void* d_out, int out_size, void* d_ws, size_t ws_size,
                              hipStream_t stream) {
    const float* x_in0 = (const float*)d_in[0];
    const int*   ei    = (const int*)d_in[1];
    const float* ea    = (const float*)d_in[2];
    const int*   batch = (const int*)d_in[3];
    const int* src = ei;
    const int* dst = ei + EE;

    // workspace layout (floats)
    float* ws = (float*)d_ws;
    const size_t NC = (size_t)NN * CD;
    float* Q   = ws;            float* Kf  = ws + NC;
    float* V   = ws + 2 * NC;   float* S   = ws + 3 * NC;
    float* AGG = ws + 4 * NC;   float* X1  = ws + 5 * NC;
    float* X2  = ws + 6 * NC;
    float* LG  = ws + 7 * NC;                       // EE*4 max
    size_t o = 7 * NC + (size_t)EE * 4;
    unsigned* MF = (unsigned*)(ws + o); o += (size_t)NN * 4;
    float* SS  = ws + o; o += (size_t)NN * 4;
    float* CS  = ws + o; o += CD;                   // colsum
    float* CQ  = ws + o; o += CD;                   // colsumsq
    float* CNT = ws + o; o += GG;
    float* PAD = ws + o; o += (size_t)GG * CD;
    unsigned* PM = (unsigned*)(ws + o); o += (size_t)GG * CD;
    float* H0  = ws + o; o += (size_t)GG * 768;
    float* H1  = ws + o; o += (size_t)GG * 768;

    const int gemmBlocks = (NN / 16) * (CD / 16) / 8;   // 2048

    auto run_layer = [&](const float* xin, int cin, int heads, int d, int dshift,
                         float inv_sqrt_d, int pbase, float* xout) {
        const float* qW = (const float*)d_in[pbase + 0];
        const float* qb = (const float*)d_in[pbase + 1];
        const float* kW = (const float*)d_in[pbase + 2];
        const float* kb = (const float*)d_in[pbase + 3];
        const float* vW = (const float*)d_in[pbase + 4];
        const float* vb = (const float*)d_in[pbase + 5];
        const float* eW = (const float*)d_in[pbase + 6];
        const float* sW = (const float*)d_in[pbase + 7];
        const float* sb = (const float*)d_in[pbase + 8];
        const float* gg = (const float*)d_in[pbase + 9];
        const float* bt = (const float*)d_in[pbase + 10];

        gemm_wmma_f32<<<gemmBlocks, 256, 0, stream>>>(xin, qW, qb, Q,  cin, CD);
        gemm_wmma_f32<<<gemmBlocks, 256, 0, stream>>>(xin, kW, kb, Kf, cin, CD);
        gemm_wmma_f32<<<gemmBlocks, 256, 0, stream>>>(xin, vW, vb, V,  cin, CD);
        gemm_wmma_f32<<<gemmBlocks, 256, 0, stream>>>(xin, sW, sb, S,  cin, CD);

        size_t nh = (size_t)NN * heads;
        fill_u32<<<(int)((nh + 255) / 256), 256, 0, stream>>>(MF, FLIP_NEG_INF, nh);
        fill_f32<<<(int)((nh + 255) / 256), 256, 0, stream>>>(SS, 0.f, nh);
        fill_f32<<<(int)(NC / 256), 256, 0, stream>>>(AGG, 0.f, NC);

        edge_logits_kernel<<<EE, 256, 0, stream>>>(Q, Kf, ea, eW, src, dst, LG, MF,
                                                   heads, d, inv_sqrt_d);
        int tot = EE * heads;
        edge_softmax_kernel<<<(tot + 255) / 256, 256, 0, stream>>>(LG, MF, SS, dst, heads, tot);
        edge_aggregate_kernel<<<EE, 256, 0, stream>>>(LG, SS, V, ea, eW, src, dst, AGG,
                                                      heads, dshift);

        fill_f32<<<2, 256, 0, stream>>>(CS, 0.f, (size_t)2 * CD);   // CS and CQ contiguous
        skip_stats_kernel<<<128, 256, 0, stream>>>(AGG, S, CS, CQ, NN / 128);
        bn_apply_kernel<<<NN, 256, 0, stream>>>(AGG, CS, CQ, gg, bt, xout);
    };

    // layer 1: heads=4, d=64;  layers 2,3: heads=1, d=256
    run_layer(x_in0, INC, 4, 64, 6, 0.125f,  4,  X1);
    run_layer(X1,    CD,  1, 256, 8, 0.0625f, 15, X2);
    run_layer(X2,    CD,  1, 256, 8, 0.0625f, 26, X1);

    // pooling
    fill_f32<<<1, 64, 0, stream>>>(CNT, 0.f, GG);
    fill_f32<<<GG, 256, 0, stream>>>(PAD, 0.f, (size_t)GG * CD);
    fill_u32<<<GG, 256, 0, stream>>>(PM, FLIP_NEG_INF, (size_t)GG * CD);
    pool_scatter_kernel<<<NN, 256, 0, stream>>>(X1, batch, PAD, PM, CNT);
    pool_finalize_kernel<<<GG, 256, 0, stream>>>(PAD, PM, CNT, H0);

    // MLP head
    const float* mW1 = (const float*)d_in[37];
    const float* mb1 = (const float*)d_in[38];
    const float* pa  = (const float*)d_in[39];
    const float* mW2 = (const float*)d_in[40];
    const float* mb2 = (const float*)d_in[41];
    mlp1_kernel<<<(GG * 768) / 256, 256, 0, stream>>>(H0, mW1, mb1, pa, H1);
    mlp2_kernel<<<1, 64, 0, stream>>>(H1, mW2, mb2, (float*)d_out);
}